// GaussianMixtureBasedImputation_1700807049543
// MI455X (gfx1250) — compile-verified
//
#include <hip/hip_runtime.h>
#include <hip/hip_bf16.h>
#include <math.h>

// Problem sizes (fixed by the reference)
#define B_DIM 2048
#define D_DIM 784
#define K_DIM 128

typedef __attribute__((ext_vector_type(2))) float v2f;
typedef __attribute__((ext_vector_type(8))) float v8f;

// ---------------------------------------------------------------------------
// Kernel 1: precompute per-center GEMM operand matrices (k-major = same layout
// as means/cov (K,D) row-major) + log mixture weights.
//   At[k,d] = -1/(2 c)            (multiplies mask*x^2)
//   Bt[k,d] =  m/c                (multiplies mask*x)
//   Ct[k,d] = -m^2/(2c) - 0.5*log c   (multiplies mask)
// ---------------------------------------------------------------------------
__global__ void gmm_precompute_kernel(const float* __restrict__ weights,
                                      const float* __restrict__ means,
                                      const float* __restrict__ cov,
                                      float* __restrict__ At,
                                      float* __restrict__ Bt,
                                      float* __restrict__ Ct,
                                      float* __restrict__ logw) {
  int i = blockIdx.x * blockDim.x + threadIdx.x;
  if (i < K_DIM * D_DIM) {
    float m = means[i];
    float c = cov[i];
    float inv = 1.0f / c;
    At[i] = -0.5f * inv;
    Bt[i] = m * inv;
    Ct[i] = -0.5f * m * m * inv - 0.5f * __logf(c);
  }
  if (i < K_DIM) {
    logw[i] = __logf(weights[i]);
  }
}

// ---------------------------------------------------------------------------
// Kernel 2: fused responsibilities kernel.
// Grid: 128 workgroups (one per 16-row batch strip) x 256 threads (8 wave32).
// Wave w owns the 16x16 output tile for centers [16w, 16w+16).
// dep(16x128) accumulated via V_WMMA_F32_16X16X4_F32 over D in steps of 4,
// 3 WMMAs per step (P*At + Q*Bt + R*Ct). Epilogue: LDS strip + per-row
// stabilized log-sum-exp (softmax) fully on-chip.
//
// fp32 WMMA operand layouts (ISA 7.12.2):
//   A (16x4):  lane&15 = M row;  VGPR0 = K(0 | 2 for hi half), VGPR1 = K(1 | 3)
//   B (4x16):  lane&15 = N col;  VGPR0 = K(0 | 2),              VGPR1 = K(1 | 3)
//   C/D:       VGPR r -> M = r (lanes 0-15) / r+8 (lanes 16-31), N = lane&15
// ---------------------------------------------------------------------------
__global__ __launch_bounds__(256)
void gmm_responsibilities_kernel(const float* __restrict__ data,
                                 const float* __restrict__ mask,
                                 const float* __restrict__ At,
                                 const float* __restrict__ Bt,
                                 const float* __restrict__ Ct,
                                 const float* __restrict__ logw,
                                 float* __restrict__ out) {
  __shared__ float sdep[16 * K_DIM];  // 8 KB dep strip

  const int wave = threadIdx.x >> 5;
  const int lane = threadIdx.x & 31;
  const int half = lane >> 4;   // 0: K=0,1   1: K=2,3
  const int l15  = lane & 15;
  const int m0   = blockIdx.x * 16;
  const int n0   = wave * 16;
  const int mrow = m0 + l15;    // batch row this lane supplies for A
  const int ncol = n0 + l15;    // center col this lane supplies for B

  const float* __restrict__ dptr = data + (size_t)mrow * D_DIM;
  const float* __restrict__ mptr = mask + (size_t)mrow * D_DIM;
  const float* __restrict__ aptr = At + (size_t)ncol * D_DIM;  // k-major: contiguous in d
  const float* __restrict__ bptr = Bt + (size_t)ncol * D_DIM;
  const float* __restrict__ cptr = Ct + (size_t)ncol * D_DIM;

  v8f acc = {0.f, 0.f, 0.f, 0.f, 0.f, 0.f, 0.f, 0.f};

  for (int d = 0; d < D_DIM; d += 4) {
    const int off = d + 2 * half;
    v2f xd = *(const v2f*)(dptr + off);   // data pair
    v2f md = *(const v2f*)(mptr + off);   // mask pair
    v2f bA = *(const v2f*)(aptr + off);   // B-operand pairs (contiguous, k-major)
    v2f bB = *(const v2f*)(bptr + off);
    v2f bC = *(const v2f*)(cptr + off);

    v2f Q = {md.x * xd.x, md.y * xd.y};        // mask*x
    v2f P = {Q.x * xd.x,  Q.y * xd.y};         // mask*x^2

    // acc += P*At + Q*Bt + mask*Ct   (all fp32 WMMA)
    acc = __builtin_amdgcn_wmma_f32_16x16x4_f32(false, P,  false, bA,
                                                (short)0, acc, false, false);
    acc = __builtin_amdgcn_wmma_f32_16x16x4_f32(false, Q,  false, bB,
                                                (short)0, acc, false, false);
    acc = __builtin_amdgcn_wmma_f32_16x16x4_f32(false, md, false, bC,
                                                (short)0, acc, false, false);
  }

  // Spill tile to LDS, adding log-weight and applying the reference NaN fix.
  const float lw = logw[ncol];
#pragma unroll
  for (int r = 0; r < 8; ++r) {
    int row = r + half * 8;
    float v = acc[r] + lw;
    v = (v != v) ? 0.0f : v;   // isnan -> 0, as in the torch/jax code
    sdep[row * K_DIM + ncol] = v;
  }
  __syncthreads();

  // Per-row stabilized softmax over K=128 (4 values/lane, wave32 xor-reduce).
  // Reference: dep - (log(sum(exp(dep - max) + 1e-8)) + max), then exp.
#pragma unroll
  for (int rr = 0; rr < 2; ++rr) {
    const int row = wave * 2 + rr;
    float v[4];
    float mx = -__builtin_inff();
#pragma unroll
    for (int j = 0; j < 4; ++j) {
      v[j] = sdep[row * K_DIM + lane + 32 * j];
      mx = fmaxf(mx, v[j]);
    }
#pragma unroll
    for (int s = 16; s >= 1; s >>= 1) mx = fmaxf(mx, __shfl_xor(mx, s, 32));

    float sum = 0.0f;
#pragma unroll
    for (int j = 0; j < 4; ++j) sum += __expf(v[j] - mx);
#pragma unroll
    for (int s = 16; s >= 1; s >>= 1) sum += __shfl_xor(sum, s, 32);

    const float logZ = __logf(sum + (float)K_DIM * 1e-8f) + mx;
    float* __restrict__ orow = out + (size_t)(m0 + row) * K_DIM;
#pragma unroll
    for (int j = 0; j < 4; ++j) orow[lane + 32 * j] = __expf(v[j] - logZ);
  }
}

// ---------------------------------------------------------------------------
// Host-side launch. Inputs (setup_inputs order):
//   0: data (B*D f32)  1: mask (B*D f32)  2: weights (K f32)
//   3: means (K*D f32) 4: covariances (K*D f32)
// Output: responsibilities (B*K f32).
// Workspace layout: At | Bt | Ct (K*D floats each) | logw (K floats).
// ---------------------------------------------------------------------------
extern "C" void kernel_launch(void* const* d_in, const int* in_sizes, int n_in,
                              void* d_out, int out_size, void* d_ws, size_t ws_size,
                              hipStream_t stream) {
  const float* data    = (const float*)d_in[0];
  const float* mask    = (const float*)d_in[1];
  const float* weights = (const float*)d_in[2];
  const float* means   = (const float*)d_in[3];
  const float* cov     = (const float*)d_in[4];
  float* out = (float*)d_out;

  float* At   = (float*)d_ws;
  float* Bt   = At + (size_t)K_DIM * D_DIM;
  float* Ct   = Bt + (size_t)K_DIM * D_DIM;
  float* logw = Ct + (size_t)K_DIM * D_DIM;

  const int pre_n = K_DIM * D_DIM;
  gmm_precompute_kernel<<<(pre_n + 255) / 256, 256, 0, stream>>>(
      weights, means, cov, At, Bt, Ct, logw);

  gmm_responsibilities_kernel<<<B_DIM / 16, 256, 0, stream>>>(
      data, mask, At, Bt, Ct, logw, out);
}